// TransformerTreeStack_72799695667839
// MI455X (gfx1250) — compile-verified
//
#include <hip/hip_runtime.h>
#include <hip/hip_bf16.h>

#define B_ 4096
#define L_ 512
#define D_ 128
#define I_ 256

typedef __attribute__((ext_vector_type(2))) float v2f;
typedef __attribute__((ext_vector_type(8))) float v8f;

// ---------------------------------------------------------------------------
// Kernel 1: fused Q/K/V projection  out = x @ W + b  via V_WMMA_F32_16X16X4_F32
// One wave32 per 16x16 output tile; blockIdx.z selects which projection.
//
// A (16x4 f32): lanes 0-15 -> M=lane, VGPR0/1 = K 0/1 ; lanes 16-31 -> K 2/3
// B (4x16 f32): lanes 0-15 -> N=lane, VGPR0/1 = K 0/1 ; lanes 16-31 -> K 2/3
// C/D (16x16 f32): VGPR j, lanes 0-15 -> M=j, N=lane ; lanes 16-31 -> M=j+8
// ---------------------------------------------------------------------------
__global__ __launch_bounds__(128) void proj_wmma_kernel(
    const float* __restrict__ x,
    const float* __restrict__ Wq, const float* __restrict__ bq,
    const float* __restrict__ Wk, const float* __restrict__ bk,
    const float* __restrict__ Wv, const float* __restrict__ bv,
    float* __restrict__ Qo, float* __restrict__ Ko, float* __restrict__ Vo)
{
    const int z  = blockIdx.z;
    const float* W  = (z == 0) ? Wq : (z == 1) ? Wk : Wv;
    const float* bs = (z == 0) ? bq : (z == 1) ? bk : bv;
    float*       O  = (z == 0) ? Qo : (z == 1) ? Ko : Vo;

    const int wave = threadIdx.x >> 5;
    const int lane = threadIdx.x & 31;
    const int half = lane >> 4;     // which K-pair this lane provides
    const int lm   = lane & 15;     // M (for A) / N (for B, C/D)
    const int m0 = (blockIdx.x * 4 + wave) * 16;
    const int n0 = blockIdx.y * 16;

    const float* arow = x + (size_t)(m0 + lm) * I_ + 2 * half;   // A: row m0+lm
    const float* bcol = W + (size_t)(2 * half) * D_ + n0 + lm;   // B: col n0+lm

    v8f c = {};
    #pragma unroll 4
    for (int k = 0; k < I_; k += 4) {
        v2f a, bm;
        a.x  = arow[k];                       // x[m, k + 2*half]
        a.y  = arow[k + 1];                   // x[m, k + 2*half + 1]
        bm.x = bcol[(size_t)k * D_];          // W[k + 2*half,     n]
        bm.y = bcol[(size_t)k * D_ + D_];     // W[k + 2*half + 1, n]
        c = __builtin_amdgcn_wmma_f32_16x16x4_f32(
                /*neg_a=*/false, a, /*neg_b=*/false, bm,
                /*c_mod=*/(short)0, c, /*reuse_a=*/false, /*reuse_b=*/false);
    }

    const float bias = bs[n0 + lm];           // N == lm for every acc VGPR
    #pragma unroll
    for (int j = 0; j < 8; ++j) {
        const int m = m0 + j + 8 * half;
        O[(size_t)m * D_ + n0 + lm] = c[j] + bias;
    }
}

// ---------------------------------------------------------------------------
// Kernel 2: masked stack attention, one 256-thread block (8 wave32) per batch
// element. Only reads the valid suffix [L-len, L) of K/V -> ~halves HBM bytes.
// ---------------------------------------------------------------------------
__global__ __launch_bounds__(256) void attn_kernel(
    const float* __restrict__ prev_keys,
    const float* __restrict__ prev_values,
    const float* __restrict__ Qws,
    const float* __restrict__ Kws,
    const float* __restrict__ Vws,
    const int*   __restrict__ lengths,
    float* __restrict__ out)
{
    __shared__ float s_scores[L_];
    __shared__ float s_wmax[8];
    __shared__ float s_wsum[8];
    __shared__ float s_pout[8 * D_];

    const int b    = blockIdx.x;
    const int tid  = threadIdx.x;
    const int wave = tid >> 5;
    const int lane = tid & 31;

    const int len   = lengths[b];
    const int start = L_ - len;
    const float inv_sqrt_d = 0.08838834764831845f;   // 1/sqrt(128)

    // lane l owns q[4l..4l+3]
    const float4 qv = *(const float4*)(Qws + (size_t)b * D_ + lane * 4);

    // new-slot score q . new_k  (computed redundantly per wave; lane-uniform)
    const float4 nk = *(const float4*)(Kws + (size_t)b * D_ + lane * 4);
    float pd = qv.x * nk.x + qv.y * nk.y + qv.z * nk.z + qv.w * nk.w;
    #pragma unroll
    for (int off = 16; off > 0; off >>= 1) pd += __shfl_xor(pd, off, 32);
    const float s_new = pd * inv_sqrt_d;

    // ---- pass 1: scores for valid prev rows + running max --------------------
    const float* Kb = prev_keys + (size_t)b * L_ * D_;
    float wmax = s_new;
    for (int n = start + wave; n < L_; n += 8) {
        const float* row = Kb + (size_t)n * D_ + lane * 4;
        __builtin_prefetch(row + 8 * D_, 0, 1);      // speculative stream-ahead
        float4 kv = *(const float4*)row;
        float d = qv.x * kv.x + qv.y * kv.y + qv.z * kv.z + qv.w * kv.w;
        #pragma unroll
        for (int off = 16; off > 0; off >>= 1) d += __shfl_xor(d, off, 32);
        const float sc = d * inv_sqrt_d;
        if (lane == 0) s_scores[n] = sc;
        wmax = fmaxf(wmax, sc);
    }
    if (lane == 0) s_wmax[wave] = wmax;
    __syncthreads();

    float gmax = s_wmax[0];
    #pragma unroll
    for (int w = 1; w < 8; ++w) gmax = fmaxf(gmax, s_wmax[w]);

    // ---- pass 2: exp weights, denominator, weighted V accumulation ----------
    const float* Vb = prev_values + (size_t)b * L_ * D_;
    float ax = 0.f, ay = 0.f, az = 0.f, aw = 0.f;    // lane l -> out[4l..4l+3]
    float lsum = 0.f;                                // lane-uniform within wave
    for (int n = start + wave; n < L_; n += 8) {
        const float* row = Vb + (size_t)n * D_ + lane * 4;
        __builtin_prefetch(row + 8 * D_, 0, 1);
        const float e = __expf(s_scores[n] - gmax);
        lsum += e;
        float4 vv = *(const float4*)row;
        ax += e * vv.x; ay += e * vv.y; az += e * vv.z; aw += e * vv.w;
    }
    if (wave == 0) {                                 // fold in the new slot
        const float e = __expf(s_new - gmax);
        lsum += e;
        float4 nv = *(const float4*)(Vws + (size_t)b * D_ + lane * 4);
        ax += e * nv.x; ay += e * nv.y; az += e * nv.z; aw += e * nv.w;
    }

    float4 acc = make_float4(ax, ay, az, aw);
    *(float4*)(s_pout + wave * D_ + lane * 4) = acc;
    if (lane == 0) s_wsum[wave] = lsum;
    __syncthreads();

    if (tid < D_) {
        float tot = 0.f, o = 0.f;
        #pragma unroll
        for (int w = 0; w < 8; ++w) { tot += s_wsum[w]; o += s_pout[w * D_ + tid]; }
        out[(size_t)b * D_ + tid] = o / tot;
    }
}

// ---------------------------------------------------------------------------
extern "C" void kernel_launch(void* const* d_in, const int* in_sizes, int n_in,
                              void* d_out, int out_size, void* d_ws, size_t ws_size,
                              hipStream_t stream) {
    const float* x   = (const float*)d_in[0];
    const float* pk  = (const float*)d_in[1];
    const float* pv  = (const float*)d_in[2];
    const float* Wq  = (const float*)d_in[3];
    const float* bq  = (const float*)d_in[4];
    const float* Wk  = (const float*)d_in[5];
    const float* bk  = (const float*)d_in[6];
    const float* Wv  = (const float*)d_in[7];
    const float* bv  = (const float*)d_in[8];
    const int*   len = (const int*)d_in[9];
    float* out = (float*)d_out;

    float* ws  = (float*)d_ws;                 // 3 * B * D floats = 6 MB
    float* Qws = ws;
    float* Kws = ws + (size_t)B_ * D_;
    float* Vws = ws + 2 * (size_t)B_ * D_;

    dim3 g1(B_ / 64, D_ / 16, 3);              // 64 x 8 x 3 blocks, 4 waves each
    proj_wmma_kernel<<<g1, 128, 0, stream>>>(x, Wq, bq, Wk, bk, Wv, bv,
                                             Qws, Kws, Vws);
    attn_kernel<<<B_, 256, 0, stream>>>(pk, pv, Qws, Kws, Vws, len, out);
}